// ThreePhaseTerm_49091476193862
// MI455X (gfx1250) — compile-verified
//
#include <hip/hip_runtime.h>
#include <stdint.h>

// ---------------------------------------------------------------------------
// ThreePhaseTerm RHS for MI455X (gfx1250).
// Sparse reaction network: no dense matmul -> WMMA inapplicable; the CDNA5
// feature used is the async global->LDS path (global_load_async_to_lds_b32 +
// s_wait_asynccnt) to stage y rows, plus ds_add_f32 LDS scatter.
// Optimizations this round:
//  * (alpha,beta,gamma) packed to one float4 table  -> 1 B128 gather/edge
//  * edge tuples packed to one int4 (sign in k MSB) -> 1 B128 stream load/edge
//  * MB=4 batch rows per block: each edge load/gather feeds 4 rate evals
// ---------------------------------------------------------------------------

#define TPB 256
constexpr int Bn   = 256;
constexpr int Sn   = 800;
constexpr int SP   = 808;     // padded LDS row stride (bank decorrelation)
constexpr int MB   = 4;       // batch rows per block
constexpr int NSMT = 600;
constexpr int NR   = 80000;
constexpr int E1   = 60000, E2  = 180000;
constexpr int E1G  = 15000, E2G = 45000;
constexpr int E1L  = 15000, E2L = 45000;
constexpr int CH_B = 6;       // divides 15000 / 45000
constexpr int CH_C = 12;      // divides 60000 / 180000
constexpr float LOG2E = 1.4426950408889634f;

struct Edges1 { const int *k, *r, *p, *s; };
struct Edges2 { const int *k, *ra, *rb, *p, *s; };

// ---- gfx1250 async global->LDS copy (GVS mode: SGPR base + VGPR offset) ----
__device__ __forceinline__ void async_ld_b32(const void* gbase, int byteOff,
                                             void* ldsPtr) {
  uint32_t lds = (uint32_t)(uintptr_t)ldsPtr;  // low 32 bits = LDS byte addr
  asm volatile("global_load_async_to_lds_b32 %0, %1, %2"
               :: "v"(lds), "v"(byteOff), "s"(gbase) : "memory");
}
__device__ __forceinline__ void wait_async() {
  asm volatile("s_wait_asynccnt 0x0" ::: "memory");
}
__device__ __forceinline__ float waveSum(float v) {
#pragma unroll
  for (int o = 16; o > 0; o >>= 1) v += __shfl_xor(v, o, 32);
  return v;
}
// alpha * (T/300)^beta * exp(-gamma/T), c1 = log2(T/300), c2 = log2e/T
__device__ __forceinline__ float arrh(float a, float bb, float gg,
                                      float c1, float c2) {
  return a * __builtin_amdgcn_exp2f(fmaf(bb, c1, -(gg * c2)));
}

// -------- pass 3: full RHS scatter into LDS, flush with global atomics ------
template <bool P>
__global__ __launch_bounds__(TPB)
void rhs_kernel(const float* __restrict__ y,
                const float* __restrict__ den_gas,
                const float* __restrict__ T_gas,
                const float* __restrict__ A, const float* __restrict__ Bc,
                const float* __restrict__ G,
                Edges1 e1, Edges2 e2,
                const int4* __restrict__ pkE1, const int4* __restrict__ pkE2,
                const float4* __restrict__ pkC,
                const float* __restrict__ smt,
                float* __restrict__ out)
{
  __shared__ float sh_y[MB][SP];
  __shared__ float sh_out[MB][SP];
  const int ch = blockIdx.x, tid = threadIdx.x;
  const int b0 = blockIdx.y * MB;

#pragma unroll
  for (int j = 0; j < MB; ++j) {
    const float* yrow = y + (size_t)(b0 + j) * Sn;
    for (int i = tid; i < Sn; i += TPB) {
      sh_out[j][i] = 0.0f;
      async_ld_b32(yrow, i * 4, &sh_y[j][i]);
    }
  }
  wait_async();
  __syncthreads();

  float c1[MB], c2[MB], dn[MB], ks2m[MB], km2s[MB];
#pragma unroll
  for (int j = 0; j < MB; ++j) {
    float T = T_gas[b0 + j];
    c1[j] = __builtin_amdgcn_logf(T * (1.0f / 300.0f));
    c2[j] = LOG2E / T;
    dn[j] = den_gas[b0 + j];
    ks2m[j] = smt[2 * (b0 + j) + 0];
    km2s[j] = smt[2 * (b0 + j) + 1];
  }

  {  // 1st-order edges (k may hit SMT-overwritten range [0,1200))
    const int n = E1 / CH_C, base = ch * n;
    for (int e = base + tid; e < base + n; e += TPB) {
      int k, r, p, s;
      if constexpr (P) {
        int4 v = pkE1[e];
        unsigned kk = (unsigned)v.x; s = kk >> 31; k = kk & 0x7fffffff;
        r = v.y; p = v.z;
      } else {
        k = e1.k[e]; r = e1.r[e]; p = e1.p[e]; s = e1.s[e];
      }
      if (k < 2 * NSMT) {
#pragma unroll
        for (int j = 0; j < MB; ++j) {
          float coeff = (k < NSMT) ? ks2m[j] : km2s[j];
          float rt = coeff * sh_y[j][r];
          atomicAdd(&sh_out[j][p], s ? rt : -rt);
        }
      } else {
        float ca, cb, cg;
        if constexpr (P) { float4 c = pkC[k]; ca = c.x; cb = c.y; cg = c.z; }
        else             { ca = A[k]; cb = Bc[k]; cg = G[k]; }
#pragma unroll
        for (int j = 0; j < MB; ++j) {
          float rt = arrh(ca, cb, cg, c1[j], c2[j]) * sh_y[j][r];
          atomicAdd(&sh_out[j][p], s ? rt : -rt);
        }
      }
    }
  }
  {  // 2nd-order edges (k >= 20000: always Arrhenius)
    const int n = E2 / CH_C, base = ch * n;
    for (int e = base + tid; e < base + n; e += TPB) {
      int k, ra, rb, p, s;
      if constexpr (P) {
        int4 v = pkE2[e];
        unsigned kk = (unsigned)v.x; s = kk >> 31; k = kk & 0x7fffffff;
        ra = v.y; rb = v.z; p = v.w;
      } else {
        k = e2.k[e]; ra = e2.ra[e]; rb = e2.rb[e]; p = e2.p[e]; s = e2.s[e];
      }
      float ca, cb, cg;
      if constexpr (P) { float4 c = pkC[k]; ca = c.x; cb = c.y; cg = c.z; }
      else             { ca = A[k]; cb = Bc[k]; cg = G[k]; }
#pragma unroll
      for (int j = 0; j < MB; ++j) {
        float rt = arrh(ca, cb, cg, c1[j], c2[j]) *
                   sh_y[j][ra] * sh_y[j][rb] * dn[j];
        atomicAdd(&sh_out[j][p], s ? rt : -rt);
      }
    }
  }
  __syncthreads();
#pragma unroll
  for (int j = 0; j < MB; ++j) {
    float* orow = out + (size_t)(b0 + j) * Sn;
    for (int i = tid; i < Sn; i += TPB) atomicAdd(&orow[i], sh_out[j][i]);
  }
}

// -------- pass 1: surface sums + masked gain/loss reductions ----------------
template <bool P>
__global__ __launch_bounds__(TPB)
void gainloss_kernel(const float* __restrict__ y,
                     const float* __restrict__ den_gas,
                     const float* __restrict__ T_gas,
                     const float* __restrict__ A, const float* __restrict__ Bc,
                     const float* __restrict__ G,
                     Edges1 g1, Edges2 g2, Edges1 l1, Edges2 l2,
                     const int4* __restrict__ pkG1, const int4* __restrict__ pkG2,
                     const int4* __restrict__ pkL1, const int4* __restrict__ pkL2,
                     const float4* __restrict__ pkC,
                     float* __restrict__ acc)   // [B,4]: ysurf,ymant,gain,loss
{
  __shared__ float sh_y[MB][SP];
  const int ch = blockIdx.x, tid = threadIdx.x;
  const int b0 = blockIdx.y * MB;

#pragma unroll
  for (int j = 0; j < MB; ++j) {
    const float* yrow = y + (size_t)(b0 + j) * Sn;
    for (int i = tid; i < Sn; i += TPB) async_ld_b32(yrow, i * 4, &sh_y[j][i]);
  }
  wait_async();
  __syncthreads();

  float c1[MB], c2[MB], dn[MB];
#pragma unroll
  for (int j = 0; j < MB; ++j) {
    float T = T_gas[b0 + j];
    c1[j] = __builtin_amdgcn_logf(T * (1.0f / 300.0f));
    c2[j] = LOG2E / T;
    dn[j] = den_gas[b0 + j];
  }

  if (ch == 0) {  // y_surf / y_mant
#pragma unroll
    for (int j = 0; j < MB; ++j) {
      float as = 0.f, am = 0.f;
      for (int i = tid; i < 300; i += TPB) { as += sh_y[j][200 + i]; am += sh_y[j][500 + i]; }
      as = waveSum(as); am = waveSum(am);
      if ((tid & 31) == 0) {
        atomicAdd(&acc[(b0 + j) * 4 + 0], as);
        atomicAdd(&acc[(b0 + j) * 4 + 1], am);
      }
    }
  }

  float ag[MB] = {}, al[MB] = {};

  auto add1 = [&](int k, int r, int s, float* dst) {
    float ca, cb, cg;
    if constexpr (P) { float4 c = pkC[k]; ca = c.x; cb = c.y; cg = c.z; }
    else             { ca = A[k]; cb = Bc[k]; cg = G[k]; }
#pragma unroll
    for (int j = 0; j < MB; ++j) {
      float rt = arrh(ca, cb, cg, c1[j], c2[j]) * sh_y[j][r];
      dst[j] += s ? rt : -rt;
    }
  };
  auto add2 = [&](int k, int ra, int rb, int s, float* dst) {
    float ca, cb, cg;
    if constexpr (P) { float4 c = pkC[k]; ca = c.x; cb = c.y; cg = c.z; }
    else             { ca = A[k]; cb = Bc[k]; cg = G[k]; }
#pragma unroll
    for (int j = 0; j < MB; ++j) {
      float rt = arrh(ca, cb, cg, c1[j], c2[j]) *
                 sh_y[j][ra] * sh_y[j][rb] * dn[j];
      dst[j] += s ? rt : -rt;
    }
  };

  {  // 1st-order gain
    const int n = E1G / CH_B, base = ch * n;
    for (int e = base + tid; e < base + n; e += TPB) {
      if constexpr (P) {
        int4 v = pkG1[e];
        if ((unsigned)(v.z - 200) < 300u) {
          unsigned kk = (unsigned)v.x;
          add1((int)(kk & 0x7fffffff), v.y, (int)(kk >> 31), ag);
        }
      } else {
        int p = g1.p[e];
        if ((unsigned)(p - 200) < 300u) add1(g1.k[e], g1.r[e], g1.s[e], ag);
      }
    }
  }
  {  // 2nd-order gain
    const int n = E2G / CH_B, base = ch * n;
    for (int e = base + tid; e < base + n; e += TPB) {
      if constexpr (P) {
        int4 v = pkG2[e];
        if ((unsigned)(v.w - 200) < 300u) {
          unsigned kk = (unsigned)v.x;
          add2((int)(kk & 0x7fffffff), v.y, v.z, (int)(kk >> 31), ag);
        }
      } else {
        int p = g2.p[e];
        if ((unsigned)(p - 200) < 300u) add2(g2.k[e], g2.ra[e], g2.rb[e], g2.s[e], ag);
      }
    }
  }
  {  // 1st-order loss
    const int n = E1L / CH_B, base = ch * n;
    for (int e = base + tid; e < base + n; e += TPB) {
      if constexpr (P) {
        int4 v = pkL1[e];
        if ((unsigned)(v.z - 200) < 300u) {
          unsigned kk = (unsigned)v.x;
          add1((int)(kk & 0x7fffffff), v.y, (int)(kk >> 31), al);
        }
      } else {
        int p = l1.p[e];
        if ((unsigned)(p - 200) < 300u) add1(l1.k[e], l1.r[e], l1.s[e], al);
      }
    }
  }
  {  // 2nd-order loss
    const int n = E2L / CH_B, base = ch * n;
    for (int e = base + tid; e < base + n; e += TPB) {
      if constexpr (P) {
        int4 v = pkL2[e];
        if ((unsigned)(v.w - 200) < 300u) {
          unsigned kk = (unsigned)v.x;
          add2((int)(kk & 0x7fffffff), v.y, v.z, (int)(kk >> 31), al);
        }
      } else {
        int p = l2.p[e];
        if ((unsigned)(p - 200) < 300u) add2(l2.k[e], l2.ra[e], l2.rb[e], l2.s[e], al);
      }
    }
  }

#pragma unroll
  for (int j = 0; j < MB; ++j) {
    float g = waveSum(ag[j]), l = waveSum(al[j]);
    if ((tid & 31) == 0) {
      atomicAdd(&acc[(b0 + j) * 4 + 2], g);
      atomicAdd(&acc[(b0 + j) * 4 + 3], l);
    }
  }
}

// -------- pass 2: per-batch k_s2m / k_m2s -----------------------------------
__global__ __launch_bounds__(TPB)
void smt_finalize(const float* __restrict__ acc, float* __restrict__ smt) {
  int b = blockIdx.x * TPB + threadIdx.x;
  if (b < Bn) {
    float ys = acc[b * 4 + 0], ym = acc[b * 4 + 1];
    float g  = acc[b * 4 + 2], l  = acc[b * 4 + 3];
    float nlayer = 1e-2f * (ys + ym);
    float decay  = fminf(2.0f / fmaxf(nlayer, 1e-30f), 1.0f);
    smt[2 * b + 0] = g / fmaxf(ys, 1e-30f);          // k_s2m
    smt[2 * b + 1] = l / fmaxf(ym, 1e-30f) * decay;  // k_m2s
  }
}

// -------- utility kernels ---------------------------------------------------
__global__ __launch_bounds__(TPB)
void zero_kernel(float* __restrict__ p, int n) {
  int i = blockIdx.x * TPB + threadIdx.x;
  if (i < n) p[i] = 0.0f;
}
__global__ __launch_bounds__(TPB)
void pack_coeff(const float* __restrict__ a, const float* __restrict__ b,
                const float* __restrict__ g, float4* __restrict__ dst, int n) {
  int i = blockIdx.x * TPB + threadIdx.x;
  if (i < n) dst[i] = make_float4(a[i], b[i], g[i], 0.0f);
}
__global__ __launch_bounds__(TPB)
void pack_e1(const int* __restrict__ k, const int* __restrict__ r,
             const int* __restrict__ p, const int* __restrict__ s,
             int4* __restrict__ dst, int n) {
  int i = blockIdx.x * TPB + threadIdx.x;
  if (i < n) dst[i] = make_int4(k[i] | (s[i] << 31), r[i], p[i], 0);
}
__global__ __launch_bounds__(TPB)
void pack_e2(const int* __restrict__ k, const int* __restrict__ ra,
             const int* __restrict__ rb, const int* __restrict__ p,
             const int* __restrict__ s, int4* __restrict__ dst, int n) {
  int i = blockIdx.x * TPB + threadIdx.x;
  if (i < n) dst[i] = make_int4(k[i] | (s[i] << 31), ra[i], rb[i], p[i]);
}

// ---------------------------------------------------------------------------
extern "C" void kernel_launch(void* const* d_in, const int* in_sizes, int n_in,
                              void* d_out, int out_size, void* d_ws, size_t ws_size,
                              hipStream_t stream) {
  const float* y       = (const float*)d_in[1];
  const float* den_gas = (const float*)d_in[2];
  const float* T_gas   = (const float*)d_in[3];
  const float* A  = (const float*)d_in[4];
  const float* Bc = (const float*)d_in[5];
  const float* G  = (const float*)d_in[6];
  Edges1 em1 = { (const int*)d_in[7],  (const int*)d_in[8],
                 (const int*)d_in[9],  (const int*)d_in[10] };
  Edges2 em2 = { (const int*)d_in[11], (const int*)d_in[12],
                 (const int*)d_in[13], (const int*)d_in[14], (const int*)d_in[15] };
  Edges1 g1  = { (const int*)d_in[16], (const int*)d_in[17],
                 (const int*)d_in[18], (const int*)d_in[19] };
  Edges2 g2  = { (const int*)d_in[20], (const int*)d_in[21],
                 (const int*)d_in[22], (const int*)d_in[23], (const int*)d_in[24] };
  Edges1 l1  = { (const int*)d_in[25], (const int*)d_in[26],
                 (const int*)d_in[27], (const int*)d_in[28] };
  Edges2 l2  = { (const int*)d_in[29], (const int*)d_in[30],
                 (const int*)d_in[31], (const int*)d_in[32], (const int*)d_in[33] };

  float* out = (float*)d_out;
  char*  ws  = (char*)d_ws;

  // packed scratch layout (16B-aligned blocks)
  const size_t oC  = 0;
  const size_t oE1 = oC  + (size_t)NR  * 16;
  const size_t oE2 = oE1 + (size_t)E1  * 16;
  const size_t oG1 = oE2 + (size_t)E2  * 16;
  const size_t oG2 = oG1 + (size_t)E1G * 16;
  const size_t oL1 = oG2 + (size_t)E2G * 16;
  const size_t oL2 = oL1 + (size_t)E1L * 16;
  const size_t oAcc = oL2 + (size_t)E2L * 16;
  const size_t NEED = oAcc + (size_t)Bn * 6 * 4;
  const bool packed = ws_size >= NEED;

  float* acc = packed ? (float*)(ws + oAcc) : (float*)ws;
  float* smt = acc + Bn * 4;

  zero_kernel<<<(Bn * Sn + TPB - 1) / TPB, TPB, 0, stream>>>(out, Bn * Sn);
  zero_kernel<<<(Bn * 4 + TPB - 1) / TPB, TPB, 0, stream>>>(acc, Bn * 4);

  const dim3 gridB(CH_B, Bn / MB), gridC(CH_C, Bn / MB);

  if (packed) {
    float4* pkC  = (float4*)(ws + oC);
    int4*  pkE1 = (int4*)(ws + oE1), *pkE2 = (int4*)(ws + oE2);
    int4*  pkG1 = (int4*)(ws + oG1), *pkG2 = (int4*)(ws + oG2);
    int4*  pkL1 = (int4*)(ws + oL1), *pkL2 = (int4*)(ws + oL2);
    pack_coeff<<<(NR + TPB - 1) / TPB, TPB, 0, stream>>>(A, Bc, G, pkC, NR);
    pack_e1<<<(E1  + TPB - 1) / TPB, TPB, 0, stream>>>(em1.k, em1.r, em1.p, em1.s, pkE1, E1);
    pack_e2<<<(E2  + TPB - 1) / TPB, TPB, 0, stream>>>(em2.k, em2.ra, em2.rb, em2.p, em2.s, pkE2, E2);
    pack_e1<<<(E1G + TPB - 1) / TPB, TPB, 0, stream>>>(g1.k, g1.r, g1.p, g1.s, pkG1, E1G);
    pack_e2<<<(E2G + TPB - 1) / TPB, TPB, 0, stream>>>(g2.k, g2.ra, g2.rb, g2.p, g2.s, pkG2, E2G);
    pack_e1<<<(E1L + TPB - 1) / TPB, TPB, 0, stream>>>(l1.k, l1.r, l1.p, l1.s, pkL1, E1L);
    pack_e2<<<(E2L + TPB - 1) / TPB, TPB, 0, stream>>>(l2.k, l2.ra, l2.rb, l2.p, l2.s, pkL2, E2L);

    gainloss_kernel<true><<<gridB, TPB, 0, stream>>>(
        y, den_gas, T_gas, A, Bc, G, g1, g2, l1, l2,
        pkG1, pkG2, pkL1, pkL2, pkC, acc);
    smt_finalize<<<1, TPB, 0, stream>>>(acc, smt);
    rhs_kernel<true><<<gridC, TPB, 0, stream>>>(
        y, den_gas, T_gas, A, Bc, G, em1, em2, pkE1, pkE2, pkC, smt, out);
  } else {
    gainloss_kernel<false><<<gridB, TPB, 0, stream>>>(
        y, den_gas, T_gas, A, Bc, G, g1, g2, l1, l2,
        nullptr, nullptr, nullptr, nullptr, nullptr, acc);
    smt_finalize<<<1, TPB, 0, stream>>>(acc, smt);
    rhs_kernel<false><<<gridC, TPB, 0, stream>>>(
        y, den_gas, T_gas, A, Bc, G, em1, em2, nullptr, nullptr, nullptr, smt, out);
  }
}